// ObjEncoder_74526272520260
// MI455X (gfx1250) — compile-verified
//
#include <hip/hip_runtime.h>
#include <math.h>

typedef float v2f __attribute__((ext_vector_type(2)));
typedef float v4f __attribute__((ext_vector_type(4)));
typedef float v8f __attribute__((ext_vector_type(8)));

#define Dm   1024
#define Bsz  64
#define Msz  128
#define Lsz  20
#define Tsz  21   // L+1 scan steps

__device__ __forceinline__ float sigm(float x) { return 1.0f / (1.0f + __expf(-x)); }

// ---------------------------------------------------------------------------
// Generic fp32 WMMA GEMM:  C[M x N] = A[M x K] @ W[N, woff:woff+K].T
//   + bias0[n] + bias1[n]          (when accumulate == 0)
//   += C                           (when accumulate == 1)
// Block: 128 threads (4 waves). Block tile 64x64; wave w covers rows
// [by*64 + 16w, +16), cols [bx*64, +64) as 4 col-tiles of 16.
// Requires: M % 64 == 0 (grid.y = M/64), N % 64 == 0 (grid.x = N/64), K % 32 == 0.
// ---------------------------------------------------------------------------
#define TM 64
#define TN 64
#define KC 32
#define SW_LD 36   // row stride (floats) of LDS W tile; 36*4=144B keeps 16B align

__global__ __launch_bounds__(128, 1) void gemm_f32_wmma(
    const float* __restrict__ A, int lda,
    const float* __restrict__ W, int ldw, int woff,
    float* __restrict__ C, int ldc, int K,
    const float* __restrict__ bias0, const float* __restrict__ bias1,
    int accumulate)
{
    __shared__ float sW[TN * SW_LD];

    const int tid  = threadIdx.x;
    const int wave = tid >> 5;
    const int lane = tid & 31;
    const int n0   = blockIdx.x * TN;
    const int m0   = blockIdx.y * TM + wave * 16;

    const int half  = lane >> 4;      // 0: lanes 0-15, 1: lanes 16-31
    const int l16   = lane & 15;
    const int kbase = half * 2;       // A/B fragment K sub-offset per ISA layout

    v8f acc0 = {}, acc1 = {}, acc2 = {}, acc3 = {};

    const float* Arow = A + (size_t)(m0 + l16) * lda;

    // Cooperative W staging map: thread -> (row wn, 16-float k slab wk0)
    const int wn  = tid >> 1;         // 0..63
    const int wk0 = (tid & 1) * 16;   // 0 or 16

    for (int kk = 0; kk < K; kk += KC) {
        __syncthreads();
        {
            const float* wsrc = W + (size_t)(n0 + wn) * ldw + woff + kk + wk0;
            float* wdst = &sW[wn * SW_LD + wk0];
#pragma unroll
            for (int j = 0; j < 16; j += 4)
                *(v4f*)(wdst + j) = *(const v4f*)(wsrc + j);
        }
        __syncthreads();

#pragma unroll
        for (int ks = 0; ks < KC; ks += 4) {
            // A fragment: lane l -> A[m0 + (l&15)][k + kbase .. +1]
            v2f a = *(const v2f*)(Arow + kk + ks + kbase);
            const float* bb = &sW[l16 * SW_LD + ks + kbase];
            v2f b0 = *(const v2f*)(bb);
            v2f b1 = *(const v2f*)(bb + 16 * SW_LD);
            v2f b2 = *(const v2f*)(bb + 32 * SW_LD);
            v2f b3 = *(const v2f*)(bb + 48 * SW_LD);
            acc0 = __builtin_amdgcn_wmma_f32_16x16x4_f32(false, a, false, b0, (short)0, acc0, false, false);
            acc1 = __builtin_amdgcn_wmma_f32_16x16x4_f32(false, a, false, b1, (short)0, acc1, false, false);
            acc2 = __builtin_amdgcn_wmma_f32_16x16x4_f32(false, a, false, b2, (short)0, acc2, false, false);
            acc3 = __builtin_amdgcn_wmma_f32_16x16x4_f32(false, a, false, b3, (short)0, acc3, false, false);
        }
    }

    // Epilogue. C tile layout: VGPR r -> row (r + 8*half), col (l&15).
#pragma unroll
    for (int r = 0; r < 8; ++r) {
        const int m = m0 + r + half * 8;
        float* crow = C + (size_t)m * ldc + n0 + l16;
        float v0 = acc0[r], v1 = acc1[r], v2 = acc2[r], v3 = acc3[r];
        if (bias0) { v0 += bias0[n0 + l16];      v1 += bias0[n0 + 16 + l16];
                     v2 += bias0[n0 + 32 + l16]; v3 += bias0[n0 + 48 + l16]; }
        if (bias1) { v0 += bias1[n0 + l16];      v1 += bias1[n0 + 16 + l16];
                     v2 += bias1[n0 + 32 + l16]; v3 += bias1[n0 + 48 + l16]; }
        if (accumulate) { v0 += crow[0]; v1 += crow[16]; v2 += crow[32]; v3 += crow[48]; }
        crow[0] = v0; crow[16] = v1; crow[32] = v2; crow[48] = v3;
    }
}

// ---------------------------------------------------------------------------
// Small elementwise / reduction kernels
// ---------------------------------------------------------------------------
__global__ void extract_kernel(const float* __restrict__ enc,
                               float* __restrict__ gv, float* __restrict__ attc)
{
    int i = blockIdx.x * blockDim.x + threadIdx.x;
    const int total = Bsz * (Msz + 1) * Dm;
    if (i >= total) return;
    int b = i / ((Msz + 1) * Dm);
    int r = i % ((Msz + 1) * Dm);
    int row = r / Dm, d = r % Dm;
    float v = enc[i];
    if (row == 0) gv[b * Dm + d] = v;
    else          attc[((size_t)b * Msz + row - 1) * Dm + d] = v;
}

__global__ void gather_obj_kernel(const float* __restrict__ attc,
                                  const int* __restrict__ obj,
                                  const float* __restrict__ bos,
                                  float* __restrict__ objf)
{
    int i = blockIdx.x * blockDim.x + threadIdx.x;
    if (i >= Bsz * Tsz * Dm) return;
    int b = i / (Tsz * Dm);
    int r = i % (Tsz * Dm);
    int t = r / Dm, d = r % Dm;
    objf[i] = (t == 0) ? bos[d]
                       : attc[((size_t)b * Msz + obj[b * Lsz + t - 1]) * Dm + d];
}

__global__ void zero_kernel(float* __restrict__ p, int n)
{
    int i = blockIdx.x * blockDim.x + threadIdx.x;
    if (i < n) p[i] = 0.0f;
}

__global__ void concat1_kernel(const float* __restrict__ h2, const float* __restrict__ gv,
                               const float* __restrict__ objf, int t,
                               float* __restrict__ inp1)
{
    int i = blockIdx.x * blockDim.x + threadIdx.x;
    if (i >= Bsz * 3 * Dm) return;
    int b = i / (3 * Dm), r = i % (3 * Dm);
    float v;
    if (r < Dm)          v = h2[b * Dm + r];
    else if (r < 2 * Dm) v = gv[b * Dm + r - Dm];
    else                 v = objf[((size_t)b * Tsz + t) * Dm + r - 2 * Dm];
    inp1[i] = v;
}

__global__ void concat2_kernel(const float* __restrict__ att, const float* __restrict__ h1,
                               float* __restrict__ inp2)
{
    int i = blockIdx.x * blockDim.x + threadIdx.x;
    if (i >= Bsz * 2 * Dm) return;
    int b = i / (2 * Dm), r = i % (2 * Dm);
    inp2[i] = (r < Dm) ? att[b * Dm + r] : h1[b * Dm + r - Dm];
}

__global__ void lstm_cell_kernel(const float* __restrict__ gates,
                                 float* __restrict__ h, float* __restrict__ c,
                                 float* __restrict__ outp, int write_out)
{
    int i = blockIdx.x * blockDim.x + threadIdx.x;
    if (i >= Bsz * Dm) return;
    int b = i / Dm, d = i % Dm;
    const float* g = gates + (size_t)b * 4 * Dm;
    float ig = sigm(g[d]);
    float fg = sigm(g[Dm + d]);
    float gg = tanhf(g[2 * Dm + d]);
    float og = sigm(g[3 * Dm + d]);
    float cn = fg * c[i] + ig * gg;
    float hn = og * tanhf(cn);
    c[i] = cn;
    h[i] = hn;
    if (write_out) outp[(size_t)b * Tsz * Dm + d] = hn;   // outp = outputs + t*Dm
}

__global__ __launch_bounds__(256, 1) void attend_kernel(
    const float* __restrict__ p_att, const float* __restrict__ ah,
    const float* __restrict__ attc, const int* __restrict__ att_mask,
    const float* __restrict__ wa, const float* __restrict__ ba,
    float* __restrict__ att)
{
    __shared__ float sc[Msz];
    __shared__ float al[Msz];
    const int b = blockIdx.x;
    const int tid = threadIdx.x;
    const int wave = tid >> 5, lane = tid & 31;

    // scores[m] = tanh(p_att[b,m,:] + ah[b,:]) . wa + ba, masked
    for (int m = wave; m < Msz; m += 8) {
        const float* pr = p_att + ((size_t)b * Msz + m) * Dm;
        const float* ar = ah + (size_t)b * Dm;
        float p = 0.0f;
        for (int d = lane; d < Dm; d += 32)
            p += tanhf(pr[d] + ar[d]) * wa[d];
#pragma unroll
        for (int off = 16; off > 0; off >>= 1)
            p += __shfl_xor(p, off, 32);
        if (lane == 0) {
            float s = p + ba[0];
            if (att_mask[b * (Msz + 1) + 1 + m] == 0) s = -1e9f;
            sc[m] = s;
        }
    }
    __syncthreads();

    // softmax (redundant per-thread scan over 128 values)
    float mmax = -3.4e38f;
    for (int m = 0; m < Msz; ++m) mmax = fmaxf(mmax, sc[m]);
    float ssum = 0.0f;
    for (int m = 0; m < Msz; ++m) ssum += __expf(sc[m] - mmax);
    float inv = 1.0f / ssum;
    if (tid < Msz) al[tid] = __expf(sc[tid] - mmax) * inv;
    __syncthreads();

    // att[b,d] = sum_m alpha[m] * att_feats[b,m,d]
    for (int d = tid; d < Dm; d += 256) {
        float a = 0.0f;
        const float* base = attc + (size_t)b * Msz * Dm + d;
        for (int m = 0; m < Msz; ++m)
            a += al[m] * base[(size_t)m * Dm];
        att[(size_t)b * Dm + d] = a;
    }
}

__global__ void logit_kernel(const float* __restrict__ ofe,
                             const float* __restrict__ afe,
                             const float* __restrict__ eos,
                             float* __restrict__ logit)
{
    const int b = blockIdx.x, l = blockIdx.y, m = threadIdx.x;
    if (m >= Msz + 1) return;
    const float* orow = ofe + ((size_t)b * Tsz + l) * Dm;
    const float* arow = (m == 0) ? eos : afe + ((size_t)b * Msz + m - 1) * Dm;
    float acc = 0.0f;
    for (int d = 0; d < Dm; ++d) acc += orow[d] * arow[d];
    logit[((size_t)b * Tsz + l) * (Msz + 1) + m] = acc;
}

// ---------------------------------------------------------------------------
// Host-side orchestration
// ---------------------------------------------------------------------------
extern "C" void kernel_launch(void* const* d_in, const int* in_sizes, int n_in,
                              void* d_out, int out_size, void* d_ws, size_t ws_size,
                              hipStream_t stream)
{
    const float* enc   = (const float*)d_in[0];
    const int*   obj   = (const int*)  d_in[1];
    const int*   amask = (const int*)  d_in[2];
    const float* bos_w = (const float*)d_in[3];
    const float* eos_w = (const float*)d_in[4];
    const float* Wp    = (const float*)d_in[5];
    const float* bp    = (const float*)d_in[6];
    const float* Wih1  = (const float*)d_in[7];
    const float* Whh1  = (const float*)d_in[8];
    const float* bih1  = (const float*)d_in[9];
    const float* bhh1  = (const float*)d_in[10];
    const float* Wih2  = (const float*)d_in[11];
    const float* Whh2  = (const float*)d_in[12];
    const float* bih2  = (const float*)d_in[13];
    const float* bhh2  = (const float*)d_in[14];
    const float* Wah   = (const float*)d_in[15];
    const float* bah   = (const float*)d_in[16];
    const float* wa    = (const float*)d_in[17];
    const float* ba    = (const float*)d_in[18];
    const float* Wae   = (const float*)d_in[19];
    const float* bae   = (const float*)d_in[20];
    const float* Woe   = (const float*)d_in[21];
    const float* boe   = (const float*)d_in[22];
    (void)in_sizes; (void)n_in; (void)out_size; (void)ws_size;

    float* outputs = (float*)d_out;                              // (B,21,D)
    float* logit   = outputs + (size_t)Bsz * Tsz * Dm;           // (B,21,129)

    // Workspace carve-up (floats)
    float* ws = (float*)d_ws;
    size_t off = 0;
    float* gv     = ws + off; off += (size_t)Bsz * Dm;
    float* attc   = ws + off; off += (size_t)Bsz * Msz * Dm;
    float* p_att  = ws + off; off += (size_t)Bsz * Msz * Dm;
    float* objf   = ws + off; off += (size_t)Bsz * Tsz * Dm;
    float* afe    = ws + off; off += (size_t)Bsz * Msz * Dm;
    float* ofe    = ws + off; off += (size_t)Bsz * Tsz * Dm;
    float* h1     = ws + off; off += (size_t)Bsz * Dm;   // h1,c1,h2,c2 contiguous
    float* c1     = ws + off; off += (size_t)Bsz * Dm;
    float* h2     = ws + off; off += (size_t)Bsz * Dm;
    float* c2     = ws + off; off += (size_t)Bsz * Dm;
    float* gates1 = ws + off; off += (size_t)Bsz * 4 * Dm;
    float* gates2 = ws + off; off += (size_t)Bsz * 4 * Dm;
    float* inp1   = ws + off; off += (size_t)Bsz * 3 * Dm;
    float* inp2   = ws + off; off += (size_t)Bsz * 2 * Dm;
    float* ah     = ws + off; off += (size_t)Bsz * Dm;
    float* attv   = ws + off; off += (size_t)Bsz * Dm;

    const int THR = 256;
    auto blk = [](int n, int t) { return (n + t - 1) / t; };

    // --- Prep ---
    extract_kernel<<<blk(Bsz * (Msz + 1) * Dm, THR), THR, 0, stream>>>(enc, gv, attc);
    gather_obj_kernel<<<blk(Bsz * Tsz * Dm, THR), THR, 0, stream>>>(attc, obj, bos_w, objf);
    zero_kernel<<<blk(4 * Bsz * Dm, THR), THR, 0, stream>>>(h1, 4 * Bsz * Dm);

    // p_att = attc @ Wp.T + bp   (8192 x 1024 x 1024)
    gemm_f32_wmma<<<dim3(Dm / TN, (Bsz * Msz) / TM), 128, 0, stream>>>(
        attc, Dm, Wp, Dm, 0, p_att, Dm, Dm, bp, nullptr, 0);
    // afe = attc @ Wae.T + bae
    gemm_f32_wmma<<<dim3(Dm / TN, (Bsz * Msz) / TM), 128, 0, stream>>>(
        attc, Dm, Wae, Dm, 0, afe, Dm, Dm, bae, nullptr, 0);

    // --- Recurrent scan: 21 steps ---
    for (int t = 0; t < Tsz; ++t) {
        concat1_kernel<<<blk(Bsz * 3 * Dm, THR), THR, 0, stream>>>(h2, gv, objf, t, inp1);

        // gates1 = inp1 @ Wih1.T + bih1 + bhh1 ;  gates1 += h1 @ Whh1.T
        gemm_f32_wmma<<<dim3(4 * Dm / TN, Bsz / TM), 128, 0, stream>>>(
            inp1, 3 * Dm, Wih1, 3 * Dm, 0, gates1, 4 * Dm, 3 * Dm, bih1, bhh1, 0);
        gemm_f32_wmma<<<dim3(4 * Dm / TN, Bsz / TM), 128, 0, stream>>>(
            h1, Dm, Whh1, Dm, 0, gates1, 4 * Dm, Dm, nullptr, nullptr, 1);
        lstm_cell_kernel<<<blk(Bsz * Dm, THR), THR, 0, stream>>>(gates1, h1, c1, nullptr, 0);

        // ah = h1 @ Wah.T + bah
        gemm_f32_wmma<<<dim3(Dm / TN, Bsz / TM), 128, 0, stream>>>(
            h1, Dm, Wah, Dm, 0, ah, Dm, Dm, bah, nullptr, 0);
        attend_kernel<<<Bsz, 256, 0, stream>>>(p_att, ah, attc, amask, wa, ba, attv);

        concat2_kernel<<<blk(Bsz * 2 * Dm, THR), THR, 0, stream>>>(attv, h1, inp2);

        // gates2 = inp2 @ Wih2.T + bih2 + bhh2 ;  gates2 += h2 @ Whh2.T (old h2)
        gemm_f32_wmma<<<dim3(4 * Dm / TN, Bsz / TM), 128, 0, stream>>>(
            inp2, 2 * Dm, Wih2, 2 * Dm, 0, gates2, 4 * Dm, 2 * Dm, bih2, bhh2, 0);
        gemm_f32_wmma<<<dim3(4 * Dm / TN, Bsz / TM), 128, 0, stream>>>(
            h2, Dm, Whh2, Dm, 0, gates2, 4 * Dm, Dm, nullptr, nullptr, 1);
        lstm_cell_kernel<<<blk(Bsz * Dm, THR), THR, 0, stream>>>(
            gates2, h2, c2, outputs + (size_t)t * Dm, 1);
    }

    // --- Epilogue ---
    // ofe = outputs @ Woe.T + boe   (1344 x 1024 x 1024)
    gemm_f32_wmma<<<dim3(Dm / TN, (Bsz * Tsz) / TM), 128, 0, stream>>>(
        outputs, Dm, Woe, Dm, 0, ofe, Dm, Dm, boe, nullptr, 0);
    // nce_logit[b,l,m] = ofe[b,l,:] . (m==0 ? eos_w : afe[b,m-1,:])
    logit_kernel<<<dim3(Bsz, Tsz), 256, 0, stream>>>(ofe, afe, eos_w, logit);
}